// MixtureOfExperts_1065151889598
// MI455X (gfx1250) — compile-verified
//
#include <hip/hip_runtime.h>
#include <hip/hip_bf16.h>

typedef __attribute__((ext_vector_type(2))) float v2f;
typedef __attribute__((ext_vector_type(8))) float v8f;
typedef __attribute__((ext_vector_type(4))) float f4;

#define D_MODEL 1024
#define D_FF    4096
#define NE      8
#define TOKENS  4096   // B*S = 2*2048
#define RBLOCKS (TOKENS / 256)

// ---------------------------------------------------------------------------
// Router: logits = x @ Wr + br ; full softmax (importance partials), top-2
// gate weights, top-2 count partials. Deterministic serial per-expert reductions.
// ---------------------------------------------------------------------------
__global__ __launch_bounds__(256) void moe_router(
    const float* __restrict__ x, const float* __restrict__ Wr,
    const float* __restrict__ br, float* __restrict__ gate_w,
    float* __restrict__ imp_part, float* __restrict__ cnt_part)
{
    int t = blockIdx.x * 256 + threadIdx.x;   // one token per thread
    float logits[NE];
#pragma unroll
    for (int e = 0; e < NE; e++) logits[e] = br[e];
    const float* xp = x + (size_t)t * D_MODEL;
    for (int d = 0; d < D_MODEL; d++) {
        float xv = xp[d];
        const float* wr = Wr + (size_t)d * NE;
#pragma unroll
        for (int e = 0; e < NE; e++) logits[e] = fmaf(xv, wr[e], logits[e]);
    }
    // full softmax over 8 experts (for importance term)
    float mx = logits[0];
#pragma unroll
    for (int e = 1; e < NE; e++) mx = fmaxf(mx, logits[e]);
    float p[NE], s = 0.f;
#pragma unroll
    for (int e = 0; e < NE; e++) { p[e] = __expf(logits[e] - mx); s += p[e]; }
    float inv = 1.f / s;
#pragma unroll
    for (int e = 0; e < NE; e++) p[e] *= inv;
    // top-2 (earliest index wins ties, matching lax.top_k)
    int i0 = 0;
#pragma unroll
    for (int e = 1; e < NE; e++) if (logits[e] > logits[i0]) i0 = e;
    int i1 = (i0 == 0) ? 1 : 0;
#pragma unroll
    for (int e = 0; e < NE; e++) if (e != i0 && logits[e] > logits[i1]) i1 = e;
    // softmax over the two kept logits (l[i0] >= l[i1])
    float e1 = __expf(logits[i1] - logits[i0]);
    float d2 = 1.f / (1.f + e1);
    float g0 = d2, g1 = e1 * d2;
    float gw[NE];
#pragma unroll
    for (int e = 0; e < NE; e++) gw[e] = 0.f;
    gw[i0] = g0; gw[i1] = g1;
#pragma unroll
    for (int e = 0; e < NE; e++) gate_w[(size_t)t * NE + e] = gw[e];

    // deterministic block reduction of importance + counts
    __shared__ float shp[256 * NE];
    __shared__ int   shi[256 * 2];
#pragma unroll
    for (int e = 0; e < NE; e++) shp[threadIdx.x * NE + e] = p[e];
    shi[threadIdx.x * 2 + 0] = i0;
    shi[threadIdx.x * 2 + 1] = i1;
    __syncthreads();
    if (threadIdx.x < NE) {
        int e = threadIdx.x;
        float si = 0.f; int c = 0;
        for (int i = 0; i < 256; i++) {
            si += shp[i * NE + e];
            c  += (shi[i * 2] == e) + (shi[i * 2 + 1] == e);
        }
        imp_part[blockIdx.x * NE + e] = si;
        cnt_part[blockIdx.x * NE + e] = (float)c;
    }
}

__global__ void moe_aux(const float* __restrict__ imp_part,
                        const float* __restrict__ cnt_part,
                        float* __restrict__ aux_out)
{
    __shared__ float acc[NE];
    if (threadIdx.x < NE) {
        int e = threadIdx.x;
        float si = 0.f, sc = 0.f;
        for (int b = 0; b < RBLOCKS; b++) {
            si += imp_part[b * NE + e];
            sc += cnt_part[b * NE + e];
        }
        acc[e] = (si / (float)TOKENS) * (sc / (float)(TOKENS * 2));
    }
    __syncthreads();
    if (threadIdx.x == 0) {
        float s = 0.f;
        for (int e = 0; e < NE; e++) s += acc[e];
        *aux_out = s * (float)NE;
    }
}

// ---------------------------------------------------------------------------
// Tiled f32 WMMA GEMM: C = A[M,K] @ B[K,N]  (M=4096 tokens)
// 128x128 block tile, BK=16, 8 waves each owning a 16x128 slice.
// MODE 0: C = relu(acc + bias)            (GEMM1 -> H workspace)
// MODE 1: C = gate * (acc + bias)         (GEMM2, first expert: init out)
// MODE 2: C += gate * (acc + bias)        (GEMM2, later experts)
// ---------------------------------------------------------------------------
template <int MODE>
__global__ __launch_bounds__(256) void moe_gemm(
    const float* __restrict__ A, const float* __restrict__ Bm,
    const float* __restrict__ bias, const float* __restrict__ gate_w,
    float* __restrict__ C, int K, int N, int expert)
{
    const int LDA = 20;                      // padded LDS stride (floats)
    __shared__ float As[128 * LDA];          // A tile, row-major [m][k]
    __shared__ float Bs[128 * LDA];          // B tile, transposed [n][k]

    int tid   = threadIdx.x;
    int wave  = tid >> 5;
    int lane  = tid & 31;
    int lhalf = lane >> 4;                   // 0/1: k-pair select
    int lmod  = lane & 15;
    int m0 = blockIdx.y * 128;
    int n0 = blockIdx.x * 128;

    v8f acc[8];
#pragma unroll
    for (int nt = 0; nt < 8; nt++)
#pragma unroll
        for (int r = 0; r < 8; r++) acc[nt][r] = 0.f;

    int am = wave * 16 + lmod;               // A fragment row (local)

    for (int k0 = 0; k0 < K; k0 += 16) {
        // --- stage A tile: 128 rows x 16 k ---
        {
            int r = tid >> 1;
            int c = (tid & 1) * 8;
            const float* gp = A + (size_t)(m0 + r) * K + k0 + c;
            f4 v0 = *(const f4*)gp;
            f4 v1 = *(const f4*)(gp + 4);
            *(f4*)&As[r * LDA + c]     = v0;
            *(f4*)&As[r * LDA + c + 4] = v1;
        }
        // --- stage B tile transposed: Bs[n][k] ---
        {
            int krow = tid >> 5;             // 0..7 (uniform per wave)
            int n4   = (tid & 31) * 4;
#pragma unroll
            for (int kk = 0; kk < 16; kk += 8) {
                const float* gp = Bm + (size_t)(k0 + krow + kk) * N + n0 + n4;
                f4 v = *(const f4*)gp;
                Bs[(n4 + 0) * LDA + krow + kk] = v.x;
                Bs[(n4 + 1) * LDA + krow + kk] = v.y;
                Bs[(n4 + 2) * LDA + krow + kk] = v.z;
                Bs[(n4 + 3) * LDA + krow + kk] = v.w;
            }
        }
        __syncthreads();
        // --- 4 k-steps x 8 n-tiles of V_WMMA_F32_16X16X4_F32 ---
#pragma unroll
        for (int kk = 0; kk < 16; kk += 4) {
            int kb = kk + lhalf * 2;
            v2f a = *(const v2f*)&As[am * LDA + kb];
#pragma unroll
            for (int nt = 0; nt < 8; nt++) {
                v2f b = *(const v2f*)&Bs[(nt * 16 + lmod) * LDA + kb];
                acc[nt] = __builtin_amdgcn_wmma_f32_16x16x4_f32(
                    false, a, false, b, (short)0, acc[nt], false, false);
            }
        }
        __syncthreads();
    }

    // --- epilogue ---
    int mrow_base = m0 + wave * 16 + lhalf * 8;
#pragma unroll
    for (int nt = 0; nt < 8; nt++) {
        int n = n0 + nt * 16 + lmod;
        float bv = bias[n];
#pragma unroll
        for (int r = 0; r < 8; r++) {
            int mrow = mrow_base + r;
            float v = acc[nt][r] + bv;
            if (MODE == 0) {
                C[(size_t)mrow * N + n] = fmaxf(v, 0.f);
            } else {
                float g = gate_w[(size_t)mrow * NE + expert];
                float* cp = &C[(size_t)mrow * N + n];
                if (MODE == 1) *cp = g * v;
                else           *cp = *cp + g * v;
            }
        }
    }
}

// ---------------------------------------------------------------------------
extern "C" void kernel_launch(void* const* d_in, const int* in_sizes, int n_in,
                              void* d_out, int out_size, void* d_ws, size_t ws_size,
                              hipStream_t stream)
{
    const float* x  = (const float*)d_in[0];
    const float* W1 = (const float*)d_in[1];
    const float* b1 = (const float*)d_in[2];
    const float* W2 = (const float*)d_in[3];
    const float* b2 = (const float*)d_in[4];
    const float* Wr = (const float*)d_in[5];
    const float* br = (const float*)d_in[6];
    float* out = (float*)d_out;

    float* ws     = (float*)d_ws;
    float* Hbuf   = ws;                                   // 4096*4096 = 64 MB
    float* gate_w = Hbuf + (size_t)TOKENS * D_FF;         // 4096*8
    float* imp    = gate_w + (size_t)TOKENS * NE;         // 16*8
    float* cnt    = imp + RBLOCKS * NE;                   // 16*8
    float* aux    = out + (size_t)TOKENS * D_MODEL;       // last output element

    moe_router<<<RBLOCKS, 256, 0, stream>>>(x, Wr, br, gate_w, imp, cnt);
    moe_aux<<<1, 32, 0, stream>>>(imp, cnt, aux);

    dim3 g1(D_FF / 128, TOKENS / 128);      // 32 x 32
    dim3 g2(D_MODEL / 128, TOKENS / 128);   // 8 x 32
    for (int e = 0; e < NE; e++) {
        moe_gemm<0><<<g1, 256, 0, stream>>>(
            x, W1 + (size_t)e * D_MODEL * D_FF, b1 + (size_t)e * D_FF,
            nullptr, Hbuf, D_MODEL, D_FF, e);
        if (e == 0)
            moe_gemm<1><<<g2, 256, 0, stream>>>(
                Hbuf, W2 + (size_t)e * D_FF * D_MODEL, b2 + (size_t)e * D_MODEL,
                gate_w, out, D_FF, D_MODEL, e);
        else
            moe_gemm<2><<<g2, 256, 0, stream>>>(
                Hbuf, W2 + (size_t)e * D_FF * D_MODEL, b2 + (size_t)e * D_MODEL,
                gate_w, out, D_FF, D_MODEL, e);
    }
}